// BipGATEncoder_39599598469225
// MI455X (gfx1250) — compile-verified
//
#include <hip/hip_runtime.h>
#include <hip/hip_bf16.h>

#define NQc   50000
#define NMc   1000
#define NNc   51000
#define EEc   500000
#define EPc   551000      // EE + NN self loops
#define HIDc  128
#define QINc  384
#define MINc  64

typedef float v2f __attribute__((ext_vector_type(2)));
typedef float v8f __attribute__((ext_vector_type(8)));

__device__ __forceinline__ float elu_f(float x) { return x > 0.f ? x : (expf(x) - 1.f); }

// monotone float <-> signed-int mapping for atomicMax-based segment max
__device__ __forceinline__ int flip_f(float f) {
    int i = __float_as_int(f);
    return i >= 0 ? i : (i ^ 0x7fffffff);
}
__device__ __forceinline__ float unflip_i(int i) {
    return __int_as_float(i >= 0 ? i : (i ^ 0x7fffffff));
}

__device__ __forceinline__ void edge_sd(const int* __restrict__ ei, int e, int& s, int& d) {
    if (e < EEc) { s = ei[e]; d = ei[EEc + e]; }
    else         { s = e - EEc; d = s; }      // appended self loops
}

// ---------------------------------------------------------------------------
// C[M x Nn] = A[M x K] @ B[K x Nn]  (+bias) (+optional elu)
// K, Nn compile-time -> B loads use immediate offsets; A tile staged in LDS
// (16 x K floats, shared by all 8 waves of the block). One wave per 16x16
// f32 output tile via V_WMMA_F32_16X16X4_F32; block covers 128 columns.
// ---------------------------------------------------------------------------
template<int K, int Nn, int ACT, bool BIAS>
__global__ void gemm_wmma_f32(const float* __restrict__ A, const float* __restrict__ B,
                              const float* __restrict__ bias, float* __restrict__ C,
                              int M)
{
    __shared__ float As[16 * K];

    const int tid  = threadIdx.x;
    const int wave = tid >> 5;
    const int lane = tid & 31;
    const int half = lane >> 4;      // 0: lanes 0-15, 1: lanes 16-31
    const int l    = lane & 15;

    const int m0 = blockIdx.x << 4;
    const int n0 = (blockIdx.y << 7) + (wave << 4);

    // cooperative float4 staging of the block's 16-row A tile
    constexpr int TOT4 = (16 * K) / 4;           // float4 elements in tile
#pragma unroll
    for (int i = tid; i < TOT4; i += 256) {
        int r  = i / (K / 4);
        int c4 = i % (K / 4);
        int row = m0 + r; if (row >= M) row = M - 1;   // clamp; stores guarded
        ((float4*)As)[i] = ((const float4*)(A + (size_t)row * K))[c4];
    }
    __syncthreads();

    const int col   = n0 + l;
    const float* Ar = As + l * K + 2 * half;
    const float* Bc = B + (size_t)(2 * half) * Nn + col;

    v8f acc = {0.f, 0.f, 0.f, 0.f, 0.f, 0.f, 0.f, 0.f};
#pragma unroll 8
    for (int k0 = 0; k0 < K; k0 += 4) {
        v2f a, b;
        // A 16x4 layout: VGPR0 = K(2*half), VGPR1 = K(2*half+1)
        a.x = Ar[k0];
        a.y = Ar[k0 + 1];
        // B 4x16 layout mirrored: rows K(2*half), K(2*half+1); col = n0+l
        b.x = Bc[k0 * Nn];
        b.y = Bc[(k0 + 1) * Nn];
        acc = __builtin_amdgcn_wmma_f32_16x16x4_f32(false, a, false, b,
                                                    (short)0, acc, false, false);
    }

    const float bv = BIAS ? bias[col] : 0.f;
#pragma unroll
    for (int r = 0; r < 8; ++r) {
        int row = m0 + half * 8 + r;     // lanes 0-15 -> rows 0-7, 16-31 -> 8-15
        if (row < M) {
            float v = acc[r] + bv;
            if (ACT) v = elu_f(v);
            C[(size_t)row * Nn + col] = v;
        }
    }
}

// a_s[n,h] = sum_c xh[n,h,c]*asrc[h,c]; a_d likewise. One thread per (n,h).
__global__ void attn_coef(const float* __restrict__ xh, const float* __restrict__ asrc,
                          const float* __restrict__ adst, float* __restrict__ a_s,
                          float* __restrict__ a_d)
{
    int idx = blockIdx.x * blockDim.x + threadIdx.x;
    if (idx >= NNc * 4) return;
    int n = idx >> 2, hh = idx & 3;
    const float4* x4 = (const float4*)(xh + (size_t)n * 512 + hh * 128);
    const float4* s4 = (const float4*)(asrc + hh * 128);
    const float4* d4 = (const float4*)(adst + hh * 128);
    float ss = 0.f, dd = 0.f;
#pragma unroll 8
    for (int i = 0; i < 32; ++i) {
        float4 x = x4[i], s = s4[i], d = d4[i];
        ss += x.x * s.x + x.y * s.y + x.z * s.z + x.w * s.w;
        dd += x.x * d.x + x.y * d.y + x.z * d.z + x.w * d.w;
    }
    a_s[idx] = ss;
    a_d[idx] = dd;
}

__global__ void zero_f32(float* __restrict__ p, size_t n)
{
    size_t i = (size_t)blockIdx.x * blockDim.x + threadIdx.x;
    if (i < n) p[i] = 0.f;
}

__global__ void init_softmax_state(int* __restrict__ amax, float* __restrict__ denom)
{
    int i = blockIdx.x * blockDim.x + threadIdx.x;
    if (i >= NNc * 4) return;
    amax[i]  = flip_f(-__builtin_inff());
    denom[i] = 0.f;
}

// logit = leaky_relu(a_s[src]+a_d[dst]); stash logit; segment max via int atomicMax
__global__ void edge_logit_max(const int* __restrict__ ei, const float4* __restrict__ a_s,
                               const float4* __restrict__ a_d, float4* __restrict__ logit,
                               int* __restrict__ amax)
{
    int e = blockIdx.x * blockDim.x + threadIdx.x;
    if (e >= EPc) return;
    int s, d; edge_sd(ei, e, s, d);
    float4 xs = a_s[s], xd = a_d[d];
    float4 lg;
    float x;
    x = xs.x + xd.x; lg.x = x > 0.f ? x : 0.2f * x;
    x = xs.y + xd.y; lg.y = x > 0.f ? x : 0.2f * x;
    x = xs.z + xd.z; lg.z = x > 0.f ? x : 0.2f * x;
    x = xs.w + xd.w; lg.w = x > 0.f ? x : 0.2f * x;
    logit[e] = lg;
    atomicMax(&amax[d * 4 + 0], flip_f(lg.x));
    atomicMax(&amax[d * 4 + 1], flip_f(lg.y));
    atomicMax(&amax[d * 4 + 2], flip_f(lg.z));
    atomicMax(&amax[d * 4 + 3], flip_f(lg.w));
}

// ea = exp(logit - amax[dst]); denom[dst] += ea  (ea overwrites logit buffer)
__global__ void edge_exp_sum(const int* __restrict__ ei, float4* __restrict__ ea,
                             const int4* __restrict__ amax, float* __restrict__ denom)
{
    int e = blockIdx.x * blockDim.x + threadIdx.x;
    if (e >= EPc) return;
    int s, d; edge_sd(ei, e, s, d);
    (void)s;
    float4 lg = ea[e];
    int4   mi = amax[d];
    float4 v;
    v.x = expf(lg.x - unflip_i(mi.x));
    v.y = expf(lg.y - unflip_i(mi.y));
    v.z = expf(lg.z - unflip_i(mi.z));
    v.w = expf(lg.w - unflip_i(mi.w));
    ea[e] = v;
    atomicAdd(&denom[d * 4 + 0], v.x);
    atomicAdd(&denom[d * 4 + 1], v.y);
    atomicAdd(&denom[d * 4 + 2], v.z);
    atomicAdd(&denom[d * 4 + 3], v.w);
}

// accum[dst,h,c] += xh[src,h,c] * (ea / (denom[dst,h]+1e-16)). One wave per
// edge; lane moves float4 (wave streams 512B per head from xh[src]).
__global__ void edge_scatter(const int* __restrict__ ei, const float* __restrict__ xh,
                             const float4* __restrict__ ea, const float4* __restrict__ denom,
                             float* __restrict__ accum)
{
    int gid  = blockIdx.x * blockDim.x + threadIdx.x;
    int e    = gid >> 5;
    int lane = gid & 31;
    if (e >= EPc) return;
    int s, d; edge_sd(ei, e, s, d);
    float4 num = ea[e];
    float4 den = denom[d];
    float4 al;
    al.x = num.x / (den.x + 1e-16f);
    al.y = num.y / (den.y + 1e-16f);
    al.z = num.z / (den.z + 1e-16f);
    al.w = num.w / (den.w + 1e-16f);
    float alpha[4] = {al.x, al.y, al.z, al.w};
    const float4* xs = (const float4*)(xh + (size_t)s * 512);
    float* ab = accum + (size_t)d * 512;
#pragma unroll
    for (int hh = 0; hh < 4; ++hh) {
        float4 x = xs[hh * 32 + lane];
        int c = hh * 128 + lane * 4;
        float a = alpha[hh];
        atomicAdd(ab + c + 0, x.x * a);
        atomicAdd(ab + c + 1, x.y * a);
        atomicAdd(ab + c + 2, x.z * a);
        atomicAdd(ab + c + 3, x.w * a);
    }
}

// out[n,c] = mean over 4 heads + bias[c] (+elu for non-final layer)
__global__ void head_mean_bias(const float* __restrict__ accum, const float* __restrict__ bias,
                               float* __restrict__ out, int act)
{
    int idx = blockIdx.x * blockDim.x + threadIdx.x;
    if (idx >= NNc * HIDc) return;
    int n = idx >> 7, c = idx & 127;
    const float* a = accum + (size_t)n * 512;
    float v = 0.25f * (a[c] + a[128 + c] + a[256 + c] + a[384 + c]) + bias[c];
    if (act) v = elu_f(v);
    out[idx] = v;
}

// ---------------------------------------------------------------------------

static void run_gat_layer(const int* ei, const float* h_in, const float* W,
                          const float* asrc, const float* adst, const float* bias,
                          float* xh, float* accum, float* a_s, float* a_d,
                          int* amax, float* denom, float* ea, float* h_out,
                          int act, hipStream_t stream)
{
    dim3 blk(256);
    gemm_wmma_f32<HIDc, 4 * HIDc, 0, false>
        <<<dim3((NNc + 15) / 16, 4), blk, 0, stream>>>(h_in, W, nullptr, xh, NNc);
    attn_coef<<<(NNc * 4 + 255) / 256, blk, 0, stream>>>(xh, asrc, adst, a_s, a_d);
    zero_f32<<<(int)(((size_t)NNc * 512 + 255) / 256), blk, 0, stream>>>(accum, (size_t)NNc * 512);
    init_softmax_state<<<(NNc * 4 + 255) / 256, blk, 0, stream>>>(amax, denom);
    edge_logit_max<<<(EPc + 255) / 256, blk, 0, stream>>>(
        ei, (const float4*)a_s, (const float4*)a_d, (float4*)ea, amax);
    edge_exp_sum<<<(EPc + 255) / 256, blk, 0, stream>>>(
        ei, (float4*)ea, (const int4*)amax, denom);
    edge_scatter<<<(int)(((size_t)EPc * 32 + 255) / 256), blk, 0, stream>>>(
        ei, xh, (const float4*)ea, (const float4*)denom, accum);
    head_mean_bias<<<(NNc * HIDc + 255) / 256, blk, 0, stream>>>(accum, bias, h_out, act);
}

extern "C" void kernel_launch(void* const* d_in, const int* in_sizes, int n_in,
                              void* d_out, int out_size, void* d_ws, size_t ws_size,
                              hipStream_t stream)
{
    const float* h_q_raw = (const float*)d_in[0];
    const float* h_m_raw = (const float*)d_in[1];
    const int*   ei      = (const int*)  d_in[2];
    const float* Wq      = (const float*)d_in[3];
    const float* bq      = (const float*)d_in[4];
    const float* Wm      = (const float*)d_in[5];
    const float* bm      = (const float*)d_in[6];
    const float* W1      = (const float*)d_in[7];
    const float* asrc1   = (const float*)d_in[8];
    const float* adst1   = (const float*)d_in[9];
    const float* b1      = (const float*)d_in[10];
    const float* W2      = (const float*)d_in[11];
    const float* asrc2   = (const float*)d_in[12];
    const float* adst2   = (const float*)d_in[13];
    const float* b2      = (const float*)d_in[14];
    float* out = (float*)d_out;

    // workspace carve-out (floats): ~274 MB total
    float* ws    = (float*)d_ws;
    float* h     = ws;  ws += (size_t)NNc * HIDc;      // [N,128]
    float* h2    = ws;  ws += (size_t)NNc * HIDc;      // [N,128]
    float* xh    = ws;  ws += (size_t)NNc * 4 * HIDc;  // [N,512]
    float* accum = ws;  ws += (size_t)NNc * 4 * HIDc;  // [N,512]
    float* a_s   = ws;  ws += (size_t)NNc * 4;
    float* a_d   = ws;  ws += (size_t)NNc * 4;
    int*   amax  = (int*)ws; ws += (size_t)NNc * 4;
    float* denom = ws;  ws += (size_t)NNc * 4;
    float* ea    = ws;                                  // [E',4]

    dim3 blk(256);
    // input projections (+bias, elu) straight into concatenated h
    gemm_wmma_f32<QINc, HIDc, 1, true>
        <<<dim3((NQc + 15) / 16, 1), blk, 0, stream>>>(h_q_raw, Wq, bq, h, NQc);
    gemm_wmma_f32<MINc, HIDc, 1, true>
        <<<dim3((NMc + 15) / 16, 1), blk, 0, stream>>>(h_m_raw, Wm, bm,
                                                       h + (size_t)NQc * HIDc, NMc);

    // layer 1 (elu) -> h2, layer 2 (no elu) -> d_out (== [h_q_out; h_m_out] flat)
    run_gat_layer(ei, h,  W1, asrc1, adst1, b1, xh, accum, a_s, a_d, amax, denom, ea, h2,  1, stream);
    run_gat_layer(ei, h2, W2, asrc2, adst2, b2, xh, accum, a_s, a_d, amax, denom, ea, out, 0, stream);
}